// PatchFolding_13666585936260
// MI455X (gfx1250) — compile-verified
//
#include <hip/hip_runtime.h>
#include <stdint.h>

// ---------------------------------------------------------------------------
// PatchFolding via TDM: out[b,h,w,(i*4+j)*64+c] = in[b,h+i-1,w+j-1,c] (0-pad)
// B=16, H=W=128, C=64 -> out [16,128,128,1024] f32 (1 GiB stored once).
//
// MI455X plan (23.3 TB/s HBM, 192 MB L2, 320 KB LDS/WGP, wave32):
//   * block = one output row (b,h): stage the 4 needed input rows (128 KB)
//     into LDS with ONE Tensor Data Mover DMA (tensor_load_to_lds, TENSORcnt)
//   * 16x tap reuse served from LDS instead of L2 -> L2 read traffic drops
//     from ~1 GiB to ~256 MB; writes are b128 non-temporal streams
//   * each thread owns fixed (tap, c4), sweeps w: ds_load_b128 -> NT
//     global_store_b128, 512 B contiguous per wave store
// ---------------------------------------------------------------------------

typedef float        v4f __attribute__((ext_vector_type(4)));
typedef unsigned int v4u __attribute__((ext_vector_type(4)));
typedef int          v4i __attribute__((ext_vector_type(4)));
typedef int          v8i __attribute__((ext_vector_type(8)));

// Geometry: floats per image row = W*C = 8192 (32768 B); out float4 per row
// = 128 px * 256 = 32768. Block index = b*128 + h.

__global__ __launch_bounds__(256) void patchfold_tdm(
    const float* __restrict__ in, v4f* __restrict__ out) {
  extern __shared__ float smem[];  // 4 rows * 8192 floats = 131072 bytes

  const uint32_t bh = blockIdx.x;      // b*128 + h
  const uint32_t h  = bh & 127u;
  const int r0 = ((int)h - 1 < 0)   ? 0   : (int)h - 1;   // first valid input row
  const int r1 = ((int)h + 2 > 127) ? 127 : (int)h + 2;   // last  valid input row
  const uint32_t nrows = (uint32_t)(r1 - r0 + 1);         // 2..4

  // ---- wave 0: one TDM DMA of nrows x 8192 f32 from global into LDS ----
  if (threadIdx.x < 32u) {
    const uint64_t gaddr =
        (uint64_t)(uintptr_t)(in + ((bh & ~127u) + (uint32_t)r0) * 8192u);
    const uint32_t lds_base = (uint32_t)(uintptr_t)(void*)smem;
    const uint32_t slot0    = (uint32_t)(r0 - ((int)h - 1));  // 0 or 1

    // D# group 0 (ISA 8.3): count=1 | lds_addr | global_addr | type=2
    v4u g0;
    g0.x = 1u;                                   // count=1 (valid), user mode
    g0.y = lds_base + slot0 * 32768u;            // LDS byte address of tile
    g0.z = (uint32_t)gaddr;                      // global_addr[31:0]
    g0.w = (uint32_t)((gaddr >> 32) & 0x01FFFFFFull) | (2u << 30);  // [56:32]|type=2

    // D# group 1 (ISA 8.4): data_size=4B, 2D tile nrows x 8192, stride 8192
    v8i g1;
    g1[0] = 0x00020000;                          // data_size=2 (4 bytes)
    g1[1] = (int)(8192u << 16);                  // tensor_dim0.lo16<<16 (bar addr 0)
    g1[2] = (int)(nrows << 16);                  // tensor_dim0.hi16 | tensor_dim1.lo16<<16
    g1[3] = (int)(8192u << 16);                  // tensor_dim1.hi16 | tile_dim0<<16
    g1[4] = (int)nrows;                          // tile_dim1 | tile_dim2<<16 (unused)
    g1[5] = 8192;                                // tensor_dim0_stride[31:0] (elements)
    g1[6] = 0;                                   // stride0.hi | stride1.lo (unused)
    g1[7] = 0;

    v4i gz4 = (v4i)0;                            // groups 2/3 unused (2D tensor)
    v8i gz8 = (v8i)0;                            // trailing group (unused, 2D)
    __builtin_amdgcn_tensor_load_to_lds(g0, g1, gz4, gz4, gz8, 0);
    __builtin_amdgcn_s_wait_tensorcnt(0);        // DMA complete -> LDS valid
  }
  __syncthreads();                               // release consumer waves

  // ---- all 8 waves: fixed (tap,c4) per thread, sweep w = 0..127 ----
  const uint32_t tid = threadIdx.x;
  const uint32_t c4  = tid & 15u;                // float4 index within 64-ch slab
  const uint32_t tap = tid >> 4;                 // i*4 + j  (0..15)
  const uint32_t r   = tap >> 2;                 // LDS row slot = i
  const int      dj  = (int)(tap & 3u) - 1;      // j - pad_before
  const int      hi  = (int)h + (int)r - 1;      // input row for this tap
  const bool rowok   = (hi >= 0) && (hi < 128);

  const v4f* s4 = (const v4f*)smem + (r * 2048u + c4);   // 2048 float4 per row
  v4f* o4 = out + (uint64_t)bh * 32768u + tid;

#pragma unroll 4
  for (uint32_t w = 0; w < 128u; ++w) {
    const int wi = (int)w + dj;
    v4f v = (v4f)0.0f;
    if (rowok & ((uint32_t)wi < 128u)) {
      v = s4[(uint32_t)wi * 16u];                // ds_load_b128
    }
    __builtin_nontemporal_store(v, o4);          // global_store_b128 ... NT
    o4 += 256;                                   // next pixel (4 KB stride)
  }
}

extern "C" void kernel_launch(void* const* d_in, const int* in_sizes, int n_in,
                              void* d_out, int out_size, void* d_ws, size_t ws_size,
                              hipStream_t stream) {
  (void)in_sizes; (void)n_in; (void)d_ws; (void)ws_size; (void)out_size;
  const float* in = (const float*)d_in[0];
  v4f* out = (v4f*)d_out;
  // 16 images * 128 rows = 2048 blocks; 128 KB dynamic LDS per block
  patchfold_tdm<<<dim3(2048), dim3(256), 131072, stream>>>(in, out);
}